// TransformerModel_45363444580889
// MI455X (gfx1250) — compile-verified
//
#include <hip/hip_runtime.h>
#include <hip/hip_bf16.h>
#include <math.h>

#define N_LOD 6
#define N_LAYERS 6
#define D_MODEL 64
#define D_INNER 128
#define D_STATE 16
#define D_CONV 4
#define DT_RANK 4
#define VOCABSZ 4096
#define BB 4
#define LL 1024
#define EPSV 1e-5f
#define DBL_W (DT_RANK + 2*D_STATE)   // 36
#define CHUNK (LL/32)                 // 32 sequence steps per lane in the parallel scan

typedef __attribute__((ext_vector_type(2))) float v2f;
typedef __attribute__((ext_vector_type(8))) float v8f;

__device__ __forceinline__ float sigmoidf_(float x){ return 1.0f/(1.0f+__expf(-x)); }

// -------------------- f32 WMMA GEMM (1 tile/wave, N-masked): C[m,n]=sum_k A[m,k]W[n,k] --
// Used only for the padded x_proj (N=36). M%16==0, K%4==0.
__global__ void wmma_gemm_kernel(const float* __restrict__ A, const float* __restrict__ W,
                                 const float* __restrict__ bias, float* __restrict__ C,
                                 int N, int K, int tilesM, int tilesN,
                                 long sA, long sW, long sC) {
  int batch = blockIdx.y;
  A += (long)batch * sA;  W += (long)batch * sW;  C += (long)batch * sC;

  int wave = threadIdx.x >> 5;
  int tile = blockIdx.x * (blockDim.x >> 5) + wave;
  if (tile >= tilesM * tilesN) return;          // wave-uniform exit (EXEC stays all-1)
  int tm = tile / tilesN;
  int tn = tile - tm * tilesN;
  int m0 = tm << 4, n0 = tn << 4;

  int lane = threadIdx.x & 31;
  int half = lane >> 4;        // K pair {0,1} vs {2,3}
  int lm   = lane & 15;

  const float* arow = A + (size_t)(m0 + lm) * K + 2*half;
  int ncol = n0 + lm;
  float wmask = (ncol < N) ? 1.0f : 0.0f;       // zero-fill padded N columns (EXEC untouched)
  int ncl = (ncol < N) ? ncol : 0;
  const float* wrow = W + (size_t)ncl * K + 2*half;

  v8f acc = {0.f,0.f,0.f,0.f,0.f,0.f,0.f,0.f};
  for (int k = 0; k < K; k += 4) {
    v2f a, b;
    a.x = arow[k];  a.y = arow[k+1];
    b.x = wrow[k] * wmask;  b.y = wrow[k+1] * wmask;
    acc = __builtin_amdgcn_wmma_f32_16x16x4_f32(false, a, false, b, (short)0, acc, false, false);
  }

  if (ncol < N) {
    float bv = bias ? bias[ncol] : 0.0f;
    #pragma unroll
    for (int r = 0; r < 8; r++) {
      int m = m0 + r + (half << 3);
      C[(size_t)m * N + ncol] = acc[r] + bv;
    }
  }
}

// -------------------- f32 WMMA GEMM, 4 N-tiles per wave (A-fragment reuse) --------------
// Requires N % 64 == 0 (in_proj 256, out_proj 64, logits 4096). 64 WMMAs / 80 loads.
__global__ void wmma_gemm4_kernel(const float* __restrict__ A, const float* __restrict__ W,
                                  const float* __restrict__ bias, float* __restrict__ C,
                                  int N, int K, int tilesM, int quadsN,
                                  long sA, long sW, long sC) {
  int batch = blockIdx.y;
  A += (long)batch * sA;  W += (long)batch * sW;  C += (long)batch * sC;

  int wave = threadIdx.x >> 5;
  int tile = blockIdx.x * (blockDim.x >> 5) + wave;
  if (tile >= tilesM * quadsN) return;          // wave-uniform
  int tm = tile / quadsN;
  int tq = tile - tm * quadsN;
  int m0 = tm << 4, n0 = tq << 6;               // 16 rows x 64 cols per wave

  int lane = threadIdx.x & 31;
  int half = lane >> 4;
  int lm   = lane & 15;

  const float* arow = A + (size_t)(m0 + lm) * K + 2*half;
  const float* w0r  = W + (size_t)(n0 + lm) * K + 2*half;
  const float* w1r  = w0r + (size_t)16 * K;
  const float* w2r  = w0r + (size_t)32 * K;
  const float* w3r  = w0r + (size_t)48 * K;

  v8f acc[4];
  #pragma unroll
  for (int j = 0; j < 4; j++) acc[j] = (v8f){0.f,0.f,0.f,0.f,0.f,0.f,0.f,0.f};

  for (int k = 0; k < K; k += 4) {
    v2f a, b0, b1, b2, b3;
    a.x  = arow[k]; a.y  = arow[k+1];
    b0.x = w0r[k];  b0.y = w0r[k+1];
    b1.x = w1r[k];  b1.y = w1r[k+1];
    b2.x = w2r[k];  b2.y = w2r[k+1];
    b3.x = w3r[k];  b3.y = w3r[k+1];
    acc[0] = __builtin_amdgcn_wmma_f32_16x16x4_f32(false, a, false, b0, (short)0, acc[0], false, false);
    acc[1] = __builtin_amdgcn_wmma_f32_16x16x4_f32(false, a, false, b1, (short)0, acc[1], false, false);
    acc[2] = __builtin_amdgcn_wmma_f32_16x16x4_f32(false, a, false, b2, (short)0, acc[2], false, false);
    acc[3] = __builtin_amdgcn_wmma_f32_16x16x4_f32(false, a, false, b3, (short)0, acc[3], false, false);
  }

  #pragma unroll
  for (int j = 0; j < 4; j++) {
    int ncol = n0 + j*16 + lm;
    float bv = bias ? bias[ncol] : 0.0f;
    #pragma unroll
    for (int r = 0; r < 8; r++) {
      int m = m0 + r + (half << 3);
      C[(size_t)m * N + ncol] = acc[j][r] + bv;
    }
  }
}

// -------------------- embedding gather: xs[n,b,l,d] = emb[n, x[b,l], d] ------------------
__global__ void embed_kernel(const int* __restrict__ x, const float* __restrict__ emb,
                             float* __restrict__ xs) {
  int idx = blockIdx.x * blockDim.x + threadIdx.x;
  if (idx >= N_LOD*BB*LL*D_MODEL) return;
  int d = idx & (D_MODEL-1);
  int l = (idx >> 6) & (LL-1);
  int b = (idx >> 16) & (BB-1);
  int n = idx >> 18;
  int tok = x[b*LL + l];
  xs[idx] = emb[((size_t)n*VOCABSZ + tok)*D_MODEL + d];
}

// -------------------- LayerNorm over D_MODEL, per-LOD affine ----------------------------
__global__ void ln_kernel(const float* __restrict__ xs, const float* __restrict__ lw,
                          const float* __restrict__ lb, float* __restrict__ h) {
  int row = blockIdx.x * blockDim.x + threadIdx.x;
  if (row >= N_LOD*BB*LL) return;
  int lod = row / (BB*LL);
  const float* xr = xs + (size_t)row * D_MODEL;
  float s = 0.f, s2 = 0.f;
  for (int d = 0; d < D_MODEL; d++) { float v = xr[d]; s += v; s2 += v*v; }
  float m   = s  * (1.0f/D_MODEL);
  float var = s2 * (1.0f/D_MODEL) - m*m;
  float inv = rsqrtf(var + EPSV);
  const float* w  = lw + lod*D_MODEL;
  const float* bv = lb + lod*D_MODEL;
  float* hr = h + (size_t)row * D_MODEL;
  for (int d = 0; d < D_MODEL; d++) hr[d] = (xr[d]-m)*inv*w[d] + bv[d];
}

// -------------------- causal depthwise conv (kernel 4) + bias + SiLU --------------------
__global__ void conv_silu_kernel(const float* __restrict__ xz, const float* __restrict__ cw,
                                 const float* __restrict__ cb, float* __restrict__ xc) {
  int idx = blockIdx.x * blockDim.x + threadIdx.x;
  if (idx >= N_LOD*BB*LL*D_INNER) return;
  int c  = idx & (D_INNER-1);
  int l  = (idx >> 7) & (LL-1);
  int nb = idx >> 17;              // n*BB + b
  int lod = nb >> 2;
  const float* base = xz + (size_t)nb * LL * (2*D_INNER) + c;
  const float* wv = cw + (lod*D_INNER + c)*D_CONV;
  float acc = cb[lod*D_INNER + c];
  #pragma unroll
  for (int j = 0; j < D_CONV; j++) {
    int ll = l - (D_CONV-1) + j;
    if (ll >= 0) acc += base[(size_t)ll * (2*D_INNER)] * wv[j];
  }
  xc[idx] = acc * sigmoidf_(acc);
}

// -------------------- parallel selective scan: one wave32 per (lod,b,channel) -----------
// h[l] = da[l]*h[l-1] + db[l] is associative; lane t owns l in [t*32, t*32+32).
// Pass 1: per-lane chunk operator (P, B) per state.  Pass 2: wave-wide shuffle scan of
// operator composition.  Pass 3: replay chunk from scanned initial state, emit gated ys.
__global__ void scan_kernel(const float* __restrict__ dbl, const float* __restrict__ xc,
                            const float* __restrict__ xz,
                            const float* __restrict__ dtw, const float* __restrict__ dtb,
                            const float* __restrict__ alog, const float* __restrict__ dp,
                            float* __restrict__ ys) {
  int wid = (blockIdx.x * blockDim.x + threadIdx.x) >> 5;   // (n*BB+b)*128 + c
  if (wid >= N_LOD*BB*D_INNER) return;                      // wave-uniform
  int lane = threadIdx.x & 31;
  int c  = wid & (D_INNER-1);
  int nb = wid >> 7;
  int lod = nb >> 2;

  const float* dr  = dbl + (size_t)nb * LL * DBL_W;
  const float* xcr = xc  + (size_t)nb * LL * D_INNER + c;
  const float* zr  = xz  + (size_t)nb * LL * (2*D_INNER) + D_INNER + c;
  float*       yr  = ys  + (size_t)nb * LL * D_INNER + c;
  const float* wdt = dtw + (lod*D_INNER + c)*DT_RANK;
  float w0 = wdt[0], w1 = wdt[1], w2 = wdt[2], w3 = wdt[3];
  float bdt = dtb[lod*D_INNER + c];
  float dpc = dp [lod*D_INNER + c];
  const float* al = alog + (size_t)(lod*D_INNER + c)*D_STATE;

  float negA[D_STATE];
  #pragma unroll
  for (int s = 0; s < D_STATE; s++) negA[s] = -__expf(al[s]);

  int lbase = lane * CHUNK;

  // ---- pass 1: chunk-local composed operator (P = prod da, Bc = chunk solution from 0)
  float P[D_STATE], Bc[D_STATE];
  #pragma unroll
  for (int s = 0; s < D_STATE; s++) { P[s] = 1.f; Bc[s] = 0.f; }
  for (int i = 0; i < CHUNK; i++) {
    int l = lbase + i;
    const float* row = dr + (size_t)l * DBL_W;
    float raw = bdt + row[0]*w0 + row[1]*w1 + row[2]*w2 + row[3]*w3;
    float dt  = (raw > 20.f) ? raw : log1pf(__expf(raw));
    float dx  = dt * xcr[(size_t)l * D_INNER];
    #pragma unroll
    for (int s = 0; s < D_STATE; s++) {
      float da = __expf(dt * negA[s]);
      P[s]  *= da;
      Bc[s]  = da * Bc[s] + dx * row[DT_RANK + s];
    }
  }

  // ---- pass 2: inclusive wave scan of (a,b) under compose(x,y) = (ax*ay, ay*bx + by),
  //      then exclusive shift -> initial state h0 of each chunk (global h starts at 0).
  float h0[D_STATE];
  #pragma unroll
  for (int s = 0; s < D_STATE; s++) {
    float a = P[s], b = Bc[s];
    #pragma unroll
    for (int off = 1; off < 32; off <<= 1) {
      float ao = __shfl_up(a, off, 32);
      float bo = __shfl_up(b, off, 32);
      if (lane >= off) { b = a * bo + b; a = a * ao; }
    }
    float hp = __shfl_up(b, 1, 32);
    h0[s] = (lane == 0) ? 0.f : hp;
  }

  // ---- pass 3: replay chunk from h0, produce ys = (sum_s h*C + D*xc) * silu(z)
  for (int i = 0; i < CHUNK; i++) {
    int l = lbase + i;
    const float* row = dr + (size_t)l * DBL_W;
    float raw = bdt + row[0]*w0 + row[1]*w1 + row[2]*w2 + row[3]*w3;
    float dt  = (raw > 20.f) ? raw : log1pf(__expf(raw));
    float xcv = xcr[(size_t)l * D_INNER];
    float dx  = dt * xcv;
    float acc = dpc * xcv;
    #pragma unroll
    for (int s = 0; s < D_STATE; s++) {
      float da = __expf(dt * negA[s]);
      h0[s] = da * h0[s] + dx * row[DT_RANK + s];
      acc  += h0[s] * row[DT_RANK + D_STATE + s];
    }
    float zv = zr[(size_t)l * (2*D_INNER)];
    yr[(size_t)l * D_INNER] = acc * (zv * sigmoidf_(zv));
  }
}

// -------------------- xs[n] = out[n] + sum_{m<n} xs_old[m] ------------------------------
__global__ void residual_kernel(float* __restrict__ xs, const float* __restrict__ outb) {
  int t = blockIdx.x * blockDim.x + threadIdx.x;
  if (t >= BB*LL*D_MODEL) return;
  float prev = 0.f;
  #pragma unroll
  for (int n = 0; n < N_LOD; n++) {
    size_t off = (size_t)n * (BB*LL*D_MODEL) + t;
    float old = xs[off];
    xs[off] = outb[off] + prev;
    prev += old;
  }
}

extern "C" void kernel_launch(void* const* d_in, const int* in_sizes, int n_in,
                              void* d_out, int out_size, void* d_ws, size_t ws_size,
                              hipStream_t stream) {
  (void)in_sizes; (void)n_in; (void)out_size; (void)ws_size;
  const int*   x    = (const int*)  d_in[0];
  const float* emb  = (const float*)d_in[1];
  const float* lnw  = (const float*)d_in[2];
  const float* lnb  = (const float*)d_in[3];
  const float* inw  = (const float*)d_in[4];
  const float* cw   = (const float*)d_in[5];
  const float* cb   = (const float*)d_in[6];
  const float* xpw  = (const float*)d_in[7];
  const float* dtw  = (const float*)d_in[8];
  const float* dtb  = (const float*)d_in[9];
  const float* alog = (const float*)d_in[10];
  const float* dp   = (const float*)d_in[11];
  const float* ow   = (const float*)d_in[12];
  const float* fcw  = (const float*)d_in[13];
  const float* fcb  = (const float*)d_in[14];
  float* out = (float*)d_out;
  float* ws  = (float*)d_ws;

  const size_t XS = (size_t)N_LOD*BB*LL*D_MODEL;
  float* xs   = ws;
  float* h    = xs  + XS;
  float* xzb  = h   + XS;                              // (n,b,l,256)
  float* xcb  = xzb + (size_t)N_LOD*BB*LL*2*D_INNER;   // (n,b,l,128)
  float* dblb = xcb + (size_t)N_LOD*BB*LL*D_INNER;     // (n,b,l,36)
  float* ysb  = dblb+ (size_t)N_LOD*BB*LL*DBL_W;       // (n,b,l,128)
  float* outb = ysb + (size_t)N_LOD*BB*LL*D_INNER;     // (n,b,l,64)

  embed_kernel<<<(int)((XS+255)/256), 256, 0, stream>>>(x, emb, xs);

  const int ROWS = N_LOD*BB*LL;                        // 24576
  for (int l = 0; l < N_LAYERS; l++) {
    ln_kernel<<<(ROWS+255)/256, 256, 0, stream>>>(
        xs, lnw + (size_t)l*N_LOD*D_MODEL, lnb + (size_t)l*N_LOD*D_MODEL, h);

    { // in_proj: per-LOD M=4096, N=256, K=64 (4 N-tiles per wave)
      int tM = (BB*LL)/16, qN = (2*D_INNER)/64;
      dim3 g((tM*qN + 7)/8, N_LOD);
      wmma_gemm4_kernel<<<g, 256, 0, stream>>>(
          h, inw + (size_t)l*N_LOD*2*D_INNER*D_MODEL, nullptr, xzb,
          2*D_INNER, D_MODEL, tM, qN,
          (long)BB*LL*D_MODEL, (long)2*D_INNER*D_MODEL, (long)BB*LL*2*D_INNER);
    }

    conv_silu_kernel<<<(N_LOD*BB*LL*D_INNER+255)/256, 256, 0, stream>>>(
        xzb, cw + (size_t)l*N_LOD*D_INNER*D_CONV, cb + (size_t)l*N_LOD*D_INNER, xcb);

    { // x_proj: per-LOD M=4096, N=36 (padded to 48), K=128 (masked 1-tile kernel)
      int tM = (BB*LL)/16, tN = (DBL_W + 15)/16;
      dim3 g((tM*tN + 7)/8, N_LOD);
      wmma_gemm_kernel<<<g, 256, 0, stream>>>(
          xcb, xpw + (size_t)l*N_LOD*DBL_W*D_INNER, nullptr, dblb,
          DBL_W, D_INNER, tM, tN,
          (long)BB*LL*D_INNER, (long)DBL_W*D_INNER, (long)BB*LL*DBL_W);
    }

    // parallel scan: one wave per (lod,b,c) -> 3072 waves
    scan_kernel<<<(N_LOD*BB*D_INNER*32 + 255)/256, 256, 0, stream>>>(
        dblb, xcb, xzb,
        dtw  + (size_t)l*N_LOD*D_INNER*DT_RANK,
        dtb  + (size_t)l*N_LOD*D_INNER,
        alog + (size_t)l*N_LOD*D_INNER*D_STATE,
        dp   + (size_t)l*N_LOD*D_INNER, ysb);

    { // out_proj: per-LOD M=4096, N=64, K=128 (4 N-tiles per wave)
      int tM = (BB*LL)/16, qN = D_MODEL/64;
      dim3 g((tM*qN + 7)/8, N_LOD);
      wmma_gemm4_kernel<<<g, 256, 0, stream>>>(
          ysb, ow + (size_t)l*N_LOD*D_MODEL*D_INNER, nullptr, outb,
          D_MODEL, D_INNER, tM, qN,
          (long)BB*LL*D_INNER, (long)D_MODEL*D_INNER, (long)BB*LL*D_MODEL);
    }

    residual_kernel<<<(BB*LL*D_MODEL+255)/256, 256, 0, stream>>>(xs, outb);
  }

  { // logits: M=24576, N=4096, K=64, +fc_b (4 N-tiles per wave)
    int tM = ROWS/16, qN = VOCABSZ/64;                 // 1536 x 64 quads
    dim3 g((tM*qN + 7)/8, 1);
    wmma_gemm4_kernel<<<g, 256, 0, stream>>>(
        xs, fcw, fcb, out, VOCABSZ, D_MODEL, tM, qN, 0, 0, 0);
  }
}